// DiscreteContinuousConvTransposeS2_46402826666241
// MI455X (gfx1250) — compile-verified
//
#include <hip/hip_runtime.h>

typedef float v2f __attribute__((ext_vector_type(2)));
typedef float v8f __attribute__((ext_vector_type(8)));

#define NLAT_IN   181
#define NLON_IN   360
#define NLAT_OUT  361
#define NLON_OUT  720
#define KSIZE     3
#define CIN       16
#define COUT      16
#define NNZ       8192
#define NPIX      (NLAT_IN * NLON_IN)      // 65160 = 4072*16 + 8
#define ACC_PITCH 727                      // odd pitch -> conflict-free co-strided LDS atomics
#define SCAT_THREADS 512

// ---------------------------------------------------------------------------
// Kernel 1: xw[k][pixel][co] = sum_ci x[ci][pixel] * weight[co][ci][k]
// One wave per 16-pixel tile; fp32 WMMA 16x16x4, K=16 in 4 chained steps,
// 3 basis tiles (kt) reusing the same A fragments. Full tiles take a uniform
// scalar branch to straight-line clause'd stores; only the single ragged tail
// tile pays per-element predication.
// ---------------------------------------------------------------------------
__global__ __launch_bounds__(32)
void disco_mix_wmma(const float* __restrict__ x,
                    const float* __restrict__ w,
                    float* __restrict__ xw)
{
    const int lane = threadIdx.x;            // 0..31 (wave32)
    const int m0   = blockIdx.x * 16;        // pixel tile base
    const int mrow = lane & 15;              // A: M = lane%16 ; B/D: N = co = lane%16
    const int half = lane >> 4;              // selects K-pair {0,1} vs {2,3} in A/B frags
    const int co   = mrow;
    const bool full_tile = (m0 + 16 <= NPIX);   // uniform across the wave

    int m = m0 + mrow;
    if (m >= NPIX) m = NPIX - 1;             // clamp loads; tail stores are predicated

    // A fragments for all 4 K-steps (Cin = 16 = 4 * 4)
    v2f a[4];
#pragma unroll
    for (int s = 0; s < 4; ++s) {
        const int ci = 4 * s + 2 * half;
        a[s].x = x[(ci + 0) * NPIX + m];
        a[s].y = x[(ci + 1) * NPIX + m];
    }

#pragma unroll
    for (int kt = 0; kt < KSIZE; ++kt) {
        v8f c = {0.f, 0.f, 0.f, 0.f, 0.f, 0.f, 0.f, 0.f};
#pragma unroll
        for (int s = 0; s < 4; ++s) {
            const int ci = 4 * s + 2 * half;
            v2f b;
            b.x = w[co * (CIN * KSIZE) + (ci + 0) * KSIZE + kt];
            b.y = w[co * (CIN * KSIZE) + (ci + 1) * KSIZE + kt];
            // (neg_a, A, neg_b, B, c_mod, C, reuse_a, reuse_b)
            c = __builtin_amdgcn_wmma_f32_16x16x4_f32(false, a[s], false, b,
                                                      (short)0, c, false, false);
        }
        // D: lane holds N=co; VGPR v holds M = v + 8*half.
        // Base address for this lane's first row; rows are 16 floats apart.
        float* dst = xw + ((size_t)(kt * NPIX + m0 + 8 * half) * COUT + co);
        if (full_tile) {
#pragma unroll
            for (int v = 0; v < 8; ++v)
                dst[v * COUT] = c[v];        // immediate-offset b32 stores, clause-able
        } else {
#pragma unroll
            for (int v = 0; v < 8; ++v) {
                const int pix = m0 + v + 8 * half;
                if (pix < NPIX) dst[v * COUT] = c[v];
            }
        }
    }
}

// ---------------------------------------------------------------------------
// Kernel 2: one workgroup per output latitude. Filter the COO list into an
// LDS queue, accumulate shifted longitude rings in an LDS fp32 accumulator
// with ds_add_f32 atomics, then write out coalesced (+bias).
// ---------------------------------------------------------------------------
__global__ __launch_bounds__(SCAT_THREADS)
void disco_scatter_lat(const float* __restrict__ xw,
                       const float* __restrict__ bias,
                       const int*   __restrict__ ker_idx,
                       const int*   __restrict__ row_idx,
                       const int*   __restrict__ col_idx,
                       const float* __restrict__ vals,
                       float* __restrict__ out)
{
    __shared__ float acc[COUT * ACC_PITCH];  // 46.5 KB ring accumulator
    __shared__ int   list[NNZ];              // worst-case work queue
    __shared__ int   cnt;

    const int tid = threadIdx.x;
    const int hi  = blockIdx.x;              // this block's output latitude

    if (tid == 0) cnt = 0;
    for (int i = tid; i < COUT * ACC_PITCH; i += SCAT_THREADS) acc[i] = 0.f;
    __syncthreads();

    // Build per-latitude nonzero queue (ds_add_rtn_u32 append)
    for (int nz = tid; nz < NNZ; nz += SCAT_THREADS) {
        if (col_idx[nz] / NLON_OUT == hi) {
            const int slot = atomicAdd(&cnt, 1);
            list[slot] = nz;
        }
    }
    __syncthreads();

    const int n = cnt;
    for (int j = 0; j < n; ++j) {
        const int   nz = list[j];
        const int   k  = ker_idx[nz];
        const int   r  = row_idx[nz];
        const int   p  = col_idx[nz] - hi * NLON_OUT;   // longitude offset
        const float v  = vals[nz];
        const float* plane = xw + (size_t)(k * NPIX + r * NLON_IN) * COUT; // [360][16]

        // 16 co x 360 po elements; co-fastest -> coalesced plane reads,
        // odd LDS pitch -> near conflict-free ds_add_f32
        for (int e = tid; e < COUT * NLON_IN; e += SCAT_THREADS) {
            const int co = e & 15;
            const int po = e >> 4;
            int lon = p + 2 * po;                        // pscale = 720/360 = 2
            if (lon >= NLON_OUT) lon -= NLON_OUT;
            atomicAdd(&acc[co * ACC_PITCH + lon], v * plane[po * COUT + co]);
        }
    }
    __syncthreads();

    // Coalesced writeout + bias
    for (int e = tid; e < COUT * NLON_OUT; e += SCAT_THREADS) {
        const int co = e / NLON_OUT;
        const int L  = e - co * NLON_OUT;
        out[(co * NLAT_OUT + hi) * NLON_OUT + L] = acc[co * ACC_PITCH + L] + bias[co];
    }
}

// ---------------------------------------------------------------------------
extern "C" void kernel_launch(void* const* d_in, const int* in_sizes, int n_in,
                              void* d_out, int out_size, void* d_ws, size_t ws_size,
                              hipStream_t stream)
{
    (void)in_sizes; (void)n_in; (void)out_size; (void)ws_size;

    const float* x    = (const float*)d_in[0];   // (1,16,181,360)
    const float* w    = (const float*)d_in[1];   // (16,16,3)
    const float* bias = (const float*)d_in[2];   // (16,)
    const int*   ker  = (const int*)  d_in[3];   // (8192,)
    const int*   row  = (const int*)  d_in[4];   // (8192,)
    const int*   col  = (const int*)  d_in[5];   // (8192,)
    const float* vals = (const float*)d_in[6];   // (8192,)
    float*       out  = (float*)d_out;           // (1,16,361,720)

    float* xw = (float*)d_ws;                    // 3 * 65160 * 16 * 4 B = 12.5 MB scratch

    const int tiles = (NPIX + 15) / 16;          // 4073 pixel tiles
    disco_mix_wmma<<<tiles, 32, 0, stream>>>(x, w, xw);
    disco_scatter_lat<<<NLAT_OUT, SCAT_THREADS, 0, stream>>>(xw, bias, ker, row, col,
                                                             vals, out);
}